// NCutLoss3D_79800492360290
// MI455X (gfx1250) — compile-verified
//
#include <hip/hip_runtime.h>
#include <hip/hip_bf16.h>
#include <math.h>

// ---------------------------------------------------------------------------
// NCut 3D loss for MI455X (gfx1250).
// Separable 9-tap Gaussian blur mapped onto V_WMMA_F32_16X16X4_F32:
//   W-axis : O(16x16) = A_data(16x24) * B_gauss(24x16)   (6 x K=4 WMMAs)
//   H/D    : O(16x16) = A_gauss(16x24) * B_data(24x16)   (6 x K=4 WMMAs)
// Each wave computes FOUR 16x16 tiles (4 slices) so the per-lane Gaussian
// fragment and the clamp/mask arithmetic are amortized over 4 WMMAs.
// All boundary handling is branchless (clamp + cndmask): EXEC stays all-ones.
// ---------------------------------------------------------------------------

typedef __attribute__((ext_vector_type(2))) float v2f;
typedef __attribute__((ext_vector_type(8))) float v8f;

#define DIMX 128
#define SLICE (128 * 128)
#define NVOX (128 * 128 * 128)

// Gaussian band coefficient: g[idx] = exp(-(idx-4)^2 / (2*5^2)) for idx in
// [0,8], else 0.  Arithmetic (v_exp_f32 + v_cndmask): no memory, no branches.
__device__ __forceinline__ float bandf(int idx) {
  float t = (float)(idx - 4);
  float v = __expf(-0.02f * t * t);
  return ((unsigned)idx < 9u) ? v : 0.0f;
}

__device__ __forceinline__ v8f wmma4(v2f a, v2f b, v8f c) {
  // 8 args: (neg_a, A, neg_b, B, c_mod, C, reuse_a, reuse_b)
  return __builtin_amdgcn_wmma_f32_16x16x4_f32(
      false, a, false, b, (short)0, c, false, false);
}

// --------------------------- setup kernels ---------------------------------

__global__ void init_kernel(float* acc) {
  if (threadIdx.x < 64) acc[threadIdx.x] = 0.0f;
}

// acc[0..7]  = sum(probs)        for bk = b*4+k
// acc[8..15] = sum(inputs*probs) for bk
__global__ __launch_bounds__(256) void sums_kernel(
    const float* __restrict__ labels, const float* __restrict__ inputs,
    float* __restrict__ acc) {
  int bk = blockIdx.y;  // 0..7
  int b = bk >> 2;
  const float* p = labels + (size_t)bk * NVOX;
  const float* x = inputs + (size_t)b * NVOX;
  float sp = 0.0f, sxp = 0.0f;
  size_t stride = (size_t)gridDim.x * blockDim.x;
  for (size_t i = (size_t)blockIdx.x * blockDim.x + threadIdx.x;
       i < (size_t)NVOX; i += stride) {
    float pv = p[i];
    sp += pv;
    sxp += x[i] * pv;
  }
  for (int off = 16; off > 0; off >>= 1) {
    sp += __shfl_xor(sp, off, 32);
    sxp += __shfl_xor(sxp, off, 32);
  }
  __shared__ float shp[8], shx[8];
  int wave = threadIdx.x >> 5;
  int lane = threadIdx.x & 31;
  if (lane == 0) {
    shp[wave] = sp;
    shx[wave] = sxp;
  }
  __syncthreads();
  if (threadIdx.x == 0) {
    float tp = 0.0f, tx = 0.0f;
    for (int i = 0; i < 8; ++i) {
      tp += shp[i];
      tx += shx[i];
    }
    atomicAdd(&acc[bk], tp);
    atomicAdd(&acc[8 + bk], tx);
  }
}

// w  = exp(-((x-mean)^2)^2), pw = probs*w   (per batch, class k)
__global__ __launch_bounds__(256) void weights_kernel(
    const float* __restrict__ labels, const float* __restrict__ inputs,
    const float* __restrict__ acc, float* __restrict__ fw,
    float* __restrict__ fpw, int k) {
  int b = blockIdx.y;
  int bk = b * 4 + k;
  const float invV = 1.0f / (float)NVOX;
  float mean = (acc[8 + bk] * invV) / (acc[bk] * invV + 1e-5f);
  const float* x = inputs + (size_t)b * NVOX;
  const float* p = labels + (size_t)bk * NVOX;
  float* w = fw + (size_t)b * NVOX;
  float* pw = fpw + (size_t)b * NVOX;
  size_t stride = (size_t)gridDim.x * blockDim.x;
  for (size_t i = (size_t)blockIdx.x * blockDim.x + threadIdx.x;
       i < (size_t)NVOX; i += stride) {
    float d = x[i] - mean;
    float d2 = d * d;
    float wv = __expf(-d2 * d2);
    w[i] = wv;
    pw[i] = p[i] * wv;
  }
}

// --------------------------- WMMA blur kernels ------------------------------
// One wave per 4 slices x one 16x16 tile; 8 waves per 256-thread block.
// 4096 wave-tiles total -> 512 blocks.

// Blur along W (contiguous axis). A = data rows, B = banded Gaussian.
__global__ __launch_bounds__(256) void blur_w_kernel(
    const float* __restrict__ src, float* __restrict__ dst) {
  int tile = blockIdx.x * 8 + (threadIdx.x >> 5);
  int lane = threadIdx.x & 31;
  int tw = tile & 7;
  int th = (tile >> 3) & 7;
  int dg = (tile >> 6) & 31;  // group of 4 depth slices
  int b = (tile >> 11) & 1;
  int n = lane & 15;   // A row (M) and C column (N)
  int sub = lane >> 4; // K-half select
  size_t base = ((size_t)b * DIMX + dg * 4) * SLICE;
  const float* srow = src + base + (size_t)(th * 16 + n) * DIMX;
  v8f c[4] = {{}, {}, {}, {}};
#pragma unroll
  for (int cc = 0; cc < 6; ++cc) {
    v2f g;
    g.x = bandf(cc * 4 + sub * 2 - n);
    g.y = bandf(cc * 4 + sub * 2 + 1 - n);
    int col = tw * 16 - 4 + cc * 4 + sub * 2;
    int i0 = min(max(col, 0), DIMX - 1);
    int i1 = min(max(col + 1, 0), DIMX - 1);
    bool m0 = (unsigned)col < (unsigned)DIMX;
    bool m1 = (unsigned)(col + 1) < (unsigned)DIMX;
#pragma unroll
    for (int j = 0; j < 4; ++j) {
      const float* sr = srow + (size_t)j * SLICE;
      float v0 = sr[i0];
      float v1 = sr[i1];
      v2f a;
      a.x = m0 ? v0 : 0.0f;
      a.y = m1 ? v1 : 0.0f;
      c[j] = wmma4(a, g, c[j]);
    }
  }
  float* orow = dst + base + (size_t)th * 16 * DIMX + tw * 16 + n;
#pragma unroll
  for (int j = 0; j < 4; ++j)
#pragma unroll
    for (int i = 0; i < 8; ++i)
      orow[(size_t)j * SLICE + (size_t)(i + sub * 8) * DIMX] = c[j][i];
}

// Blur along H. A = banded Gaussian, B = data columns.
__global__ __launch_bounds__(256) void blur_h_kernel(
    const float* __restrict__ src, float* __restrict__ dst) {
  int tile = blockIdx.x * 8 + (threadIdx.x >> 5);
  int lane = threadIdx.x & 31;
  int tw = tile & 7;
  int th = (tile >> 3) & 7;
  int dg = (tile >> 6) & 31;
  int b = (tile >> 11) & 1;
  int n = lane & 15;
  int sub = lane >> 4;
  int w = tw * 16 + n;
  size_t base = ((size_t)b * DIMX + dg * 4) * SLICE;
  const float* scol = src + base + w;  // index along H with stride DIMX
  v8f c[4] = {{}, {}, {}, {}};
#pragma unroll
  for (int cc = 0; cc < 6; ++cc) {
    v2f a;
    a.x = bandf(cc * 4 + sub * 2 - n);
    a.y = bandf(cc * 4 + sub * 2 + 1 - n);
    int h0 = th * 16 - 4 + cc * 4 + sub * 2;
    int i0 = min(max(h0, 0), DIMX - 1) * DIMX;
    int i1 = min(max(h0 + 1, 0), DIMX - 1) * DIMX;
    bool m0 = (unsigned)h0 < (unsigned)DIMX;
    bool m1 = (unsigned)(h0 + 1) < (unsigned)DIMX;
#pragma unroll
    for (int j = 0; j < 4; ++j) {
      const float* sc = scol + (size_t)j * SLICE;
      float v0 = sc[i0];
      float v1 = sc[i1];
      v2f bv;
      bv.x = m0 ? v0 : 0.0f;
      bv.y = m1 ? v1 : 0.0f;
      c[j] = wmma4(a, bv, c[j]);
    }
  }
  float* ocol = dst + base + (size_t)th * 16 * DIMX + w;
#pragma unroll
  for (int j = 0; j < 4; ++j)
#pragma unroll
    for (int i = 0; i < 8; ++i)
      ocol[(size_t)j * SLICE + (size_t)(i + sub * 8) * DIMX] = c[j][i];
}

// Blur along D, fused with  sum(probs * blurred)  reduction.
__global__ __launch_bounds__(256) void blur_d_reduce_kernel(
    const float* __restrict__ src, const float* __restrict__ labels, int k,
    float* __restrict__ accum) {
  int tile = blockIdx.x * 8 + (threadIdx.x >> 5);
  int lane = threadIdx.x & 31;
  int tw = tile & 7;
  int hg = (tile >> 3) & 31;  // group of 4 h rows
  int td = (tile >> 8) & 7;
  int b = (tile >> 11) & 1;
  int n = lane & 15;
  int sub = lane >> 4;
  int w = tw * 16 + n;
  const float* s = src + (size_t)b * NVOX + (size_t)hg * 4 * DIMX + w;
  const float* p = labels + ((size_t)b * 4 + k) * NVOX;
  v8f c[4] = {{}, {}, {}, {}};
#pragma unroll
  for (int cc = 0; cc < 6; ++cc) {
    v2f a;
    a.x = bandf(cc * 4 + sub * 2 - n);
    a.y = bandf(cc * 4 + sub * 2 + 1 - n);
    int d0 = td * 16 - 4 + cc * 4 + sub * 2;
    size_t i0 = (size_t)(min(max(d0, 0), DIMX - 1)) * SLICE;
    size_t i1 = (size_t)(min(max(d0 + 1, 0), DIMX - 1)) * SLICE;
    bool m0 = (unsigned)d0 < (unsigned)DIMX;
    bool m1 = (unsigned)(d0 + 1) < (unsigned)DIMX;
#pragma unroll
    for (int j = 0; j < 4; ++j) {
      const float* sc = s + (size_t)j * DIMX;  // h = hg*4 + j
      float v0 = sc[i0];
      float v1 = sc[i1];
      v2f bv;
      bv.x = m0 ? v0 : 0.0f;
      bv.y = m1 ? v1 : 0.0f;
      c[j] = wmma4(a, bv, c[j]);
    }
  }
  float local = 0.0f;
  const float* pb = p + (size_t)hg * 4 * DIMX + w;
#pragma unroll
  for (int j = 0; j < 4; ++j) {
#pragma unroll
    for (int i = 0; i < 8; ++i) {
      int dd = td * 16 + i + sub * 8;
      local += c[j][i] * pb[(size_t)dd * SLICE + (size_t)j * DIMX];
    }
  }
  for (int off = 16; off > 0; off >>= 1) local += __shfl_xor(local, off, 32);
  if (lane == 0) atomicAdd(accum, local);
}

__global__ void finalize_kernel(const float* __restrict__ acc,
                                float* __restrict__ out) {
  float loss = 0.0f;
  for (int k = 0; k < 4; ++k)
    loss += fabsf(acc[16 + k] / (acc[20 + k] + 1e-6f));
  out[0] = 4.0f - loss;
}

// ----------------------------- launch --------------------------------------

extern "C" void kernel_launch(void* const* d_in, const int* in_sizes, int n_in,
                              void* d_out, int out_size, void* d_ws,
                              size_t ws_size, hipStream_t stream) {
  const float* labels = (const float*)d_in[0]; // (2,4,128,128,128) f32
  const float* inputs = (const float*)d_in[1]; // (2,1,128,128,128) f32
  float* out = (float*)d_out;
  float* ws = (float*)d_ws;

  // ws layout: [0..63] accumulators | fw (2V) | fpw (2V) | tmp (2V)
  float* acc = ws;
  float* fw = ws + 64;
  float* fpw = fw + 2 * (size_t)NVOX;
  float* tmp = fpw + 2 * (size_t)NVOX;

  init_kernel<<<1, 64, 0, stream>>>(acc);
  sums_kernel<<<dim3(256, 8), 256, 0, stream>>>(labels, inputs, acc);

  for (int k = 0; k < 4; ++k) {
    weights_kernel<<<dim3(512, 2), 256, 0, stream>>>(labels, inputs, acc, fw,
                                                     fpw, k);
    // numerator: blur3d(probs*weights) . probs
    blur_w_kernel<<<512, 256, 0, stream>>>(fpw, tmp);
    blur_h_kernel<<<512, 256, 0, stream>>>(tmp, fpw);
    blur_d_reduce_kernel<<<512, 256, 0, stream>>>(fpw, labels, k,
                                                  acc + 16 + k);
    // denominator: blur3d(weights) . probs
    blur_w_kernel<<<512, 256, 0, stream>>>(fw, tmp);
    blur_h_kernel<<<512, 256, 0, stream>>>(tmp, fw);
    blur_d_reduce_kernel<<<512, 256, 0, stream>>>(fw, labels, k,
                                                  acc + 20 + k);
  }
  finalize_kernel<<<1, 1, 0, stream>>>(acc, out);
}